// ESSAttn_31026843746548
// MI455X (gfx1250) — compile-verified
//
#include <hip/hip_runtime.h>
#include <hip/hip_bf16.h>
#include <math.h>

typedef float v2f __attribute__((ext_vector_type(2)));
typedef float v8f __attribute__((ext_vector_type(8)));

#define B_   8
#define C_   512
#define C3_  1536
#define N_   4096
#define EPS_DIV  1e-7f
#define EPS_NORM 1e-12f
#define INV_SQRT_N 0.015625f   // 1/64

static __device__ __forceinline__ v8f wmma4(v2f a, v2f b, v8f c) {
    // D = A(16x4 f32) x B(4x16 f32) + C(16x16 f32)
    return __builtin_amdgcn_wmma_f32_16x16x4_f32(
        /*neg_a=*/false, a, /*neg_b=*/false, b,
        /*c_mod=*/(short)0, c, /*reuse_a=*/false, /*reuse_b=*/false);
}

// ---------------------------------------------------------------------------
// K1: qkv = x^T @ Wqkv + bqkv ; split into q,k,v buffers [B,N,C]
// x is [B,C,N] so A = x^T is K-major in memory (coalesced A loads).
// One wave computes a 32(M=n) x 64(J) tile: 2x4 register blocking,
// 8 WMMAs per 12 scalar loads per K=4 step.
// ---------------------------------------------------------------------------
__global__ void qkv_gemm(const float* __restrict__ x,
                         const float* __restrict__ Wqkv,
                         const float* __restrict__ bqkv,
                         float* __restrict__ q,
                         float* __restrict__ k,
                         float* __restrict__ v) {
    const int wave  = blockIdx.x * (blockDim.x >> 5) + (threadIdx.x >> 5);
    const int lane  = threadIdx.x & 31;
    const int laneN = lane & 15;
    const int half  = lane >> 4;

    const int JT = C3_ / 64;      // 24
    const int MT = N_ / 32;       // 128 (pairs of 16-row tiles)
    const int b  = wave / (MT * JT);
    const int rem = wave % (MT * JT);
    const int m0 = (rem / JT) * 32;
    const int j0 = (rem % JT) * 64;

    const float* xb = x + (size_t)b * C_ * N_;

    v8f acc[2][4] = {};
#pragma unroll 2
    for (int k0 = 0; k0 < C_; k0 += 4) {
        const int kk = k0 + 2 * half;
        const float* x0 = xb + (size_t)kk * N_ + m0 + laneN;
        const float* x1 = x0 + N_;
        v2f a[2];
        a[0].x = x0[0];  a[0].y = x1[0];
        a[1].x = x0[16]; a[1].y = x1[16];
        const float* w0 = Wqkv + (size_t)kk * C3_ + j0 + laneN;
        const float* w1 = w0 + C3_;
#pragma unroll
        for (int t = 0; t < 4; ++t) {
            v2f bf;
            bf.x = w0[t * 16];
            bf.y = w1[t * 16];
            acc[0][t] = wmma4(a[0], bf, acc[0][t]);
            acc[1][t] = wmma4(a[1], bf, acc[1][t]);
        }
    }

#pragma unroll
    for (int t = 0; t < 4; ++t) {
        const int j    = j0 + t * 16 + laneN;
        const float bb = bqkv[j];
        const int sec  = j >> 9;           // 0:q 1:k 2:v
        const int jj   = j & 511;
        float* dst = (sec == 0) ? q : (sec == 1) ? k : v;
        dst += (size_t)b * N_ * C_ + jj;
#pragma unroll
        for (int mt = 0; mt < 2; ++mt) {
#pragma unroll
            for (int r = 0; r < 8; ++r) {
                const int m = m0 + 16 * mt + r + 8 * half;
                dst[(size_t)m * C_] = acc[mt][t][r] + bb;
            }
        }
    }
}

// ---------------------------------------------------------------------------
// K2: per-row (b,n): mean-center q,k ; q2n (incl. axis=-1 L2 normalize) over q;
// k2_tmp = kc^2/(k2s+eps) over k (column normalize deferred to kv epilogue).
// ---------------------------------------------------------------------------
__device__ __forceinline__ float blockReduceSum(float val, float* sbuf) {
    const int tid = threadIdx.x;
    sbuf[tid] = val;
    __syncthreads();
#pragma unroll
    for (int s = 128; s > 0; s >>= 1) {
        if (tid < s) sbuf[tid] += sbuf[tid + s];
        __syncthreads();
    }
    const float r = sbuf[0];
    __syncthreads();
    return r;
}

__global__ void rownorm(float* __restrict__ q, float* __restrict__ k) {
    __shared__ float sbuf[256];
    const int row = blockIdx.x;              // b*N + n
    const int tid = threadIdx.x;
    float* qr = q + (size_t)row * C_;
    float* kr = k + (size_t)row * C_;

    float q0 = qr[tid], q1 = qr[tid + 256];
    float k0 = kr[tid], k1 = kr[tid + 256];

    const float mq = blockReduceSum(q0 + q1, sbuf) * (1.0f / (float)C_);
    const float mk = blockReduceSum(k0 + k1, sbuf) * (1.0f / (float)C_);
    q0 -= mq; q1 -= mq; k0 -= mk; k1 -= mk;

    const float q20 = q0 * q0, q21 = q1 * q1;
    const float k20 = k0 * k0, k21 = k1 * k1;

    const float q2s = blockReduceSum(q20 + q21, sbuf);
    const float q4s = blockReduceSum(q20 * q20 + q21 * q21, sbuf);
    const float k2s = blockReduceSum(k20 + k21, sbuf);

    const float sdiv = q2s + EPS_DIV;
    // ||q2/sdiv||_2 = sqrt(sum q2^2)/sdiv
    const float nrm = fmaxf(sqrtf(q4s) / sdiv, EPS_NORM);
    const float qscale = 1.0f / (sdiv * nrm);
    qr[tid]       = q20 * qscale;
    qr[tid + 256] = q21 * qscale;

    const float kscale = 1.0f / (k2s + EPS_DIV);
    kr[tid]       = k20 * kscale;
    kr[tid + 256] = k21 * kscale;
}

// ---------------------------------------------------------------------------
// K2b: colnorm[b,c] = max(||k2_tmp[b,:,c]||_2, 1e-12)
// ---------------------------------------------------------------------------
__global__ void colnorm_kernel(const float* __restrict__ k,
                               float* __restrict__ colnorm) {
    const int b = blockIdx.x / (C_ / 256);
    const int c = (blockIdx.x % (C_ / 256)) * 256 + threadIdx.x;
    const float* kb = k + (size_t)b * N_ * C_ + c;
    float acc = 0.0f;
#pragma unroll 8
    for (int n = 0; n < N_; ++n) {
        const float vv = kb[(size_t)n * C_];
        acc += vv * vv;
    }
    colnorm[b * C_ + c] = fmaxf(sqrtf(acc), EPS_NORM);
}

// ---------------------------------------------------------------------------
// K3: kv[b,c,d] = (sum_n k2_tmp[b,n,c] * v[b,n,d]) / colnorm[b,c]
// A = k2_tmp^T (K-major across n, contiguous lanes in c), B = v.
// 2x4 register blocking: 32(c) x 64(d) per wave.
// ---------------------------------------------------------------------------
__global__ void kv_gemm(const float* __restrict__ k,
                        const float* __restrict__ v,
                        const float* __restrict__ colnorm,
                        float* __restrict__ kv) {
    const int wave  = blockIdx.x * (blockDim.x >> 5) + (threadIdx.x >> 5);
    const int lane  = threadIdx.x & 31;
    const int laneN = lane & 15;
    const int half  = lane >> 4;

    const int JT = C_ / 64;       // 8
    const int MT = C_ / 32;       // 16
    const int b  = wave / (MT * JT);
    const int rem = wave % (MT * JT);
    const int c0 = (rem / JT) * 32;
    const int d0 = (rem % JT) * 64;

    const float* kb = k + (size_t)b * N_ * C_;
    const float* vb = v + (size_t)b * N_ * C_;

    v8f acc[2][4] = {};
#pragma unroll 2
    for (int k0 = 0; k0 < N_; k0 += 4) {
        const int kk = k0 + 2 * half;
        const float* ka0 = kb + (size_t)kk * C_ + c0 + laneN;
        const float* ka1 = ka0 + C_;
        v2f a[2];
        a[0].x = ka0[0];  a[0].y = ka1[0];
        a[1].x = ka0[16]; a[1].y = ka1[16];
        const float* v0 = vb + (size_t)kk * C_ + d0 + laneN;
        const float* v1 = v0 + C_;
#pragma unroll
        for (int t = 0; t < 4; ++t) {
            v2f bf;
            bf.x = v0[t * 16];
            bf.y = v1[t * 16];
            acc[0][t] = wmma4(a[0], bf, acc[0][t]);
            acc[1][t] = wmma4(a[1], bf, acc[1][t]);
        }
    }

    float* kvb = kv + (size_t)b * C_ * C_;
#pragma unroll
    for (int mt = 0; mt < 2; ++mt) {
#pragma unroll
        for (int r = 0; r < 8; ++r) {
            const int c = c0 + 16 * mt + r + 8 * half;
            const float inv = 1.0f / colnorm[b * C_ + c];
#pragma unroll
            for (int t = 0; t < 4; ++t) {
                kvb[(size_t)c * C_ + d0 + t * 16 + laneN] = acc[mt][t][r] * inv;
            }
        }
    }
}

// ---------------------------------------------------------------------------
// K4a: attn = v + (q2n @ kv) / sqrt(N), written in-place over v.
// A loads are contiguous K pairs -> single b64 per fragment element.
// ---------------------------------------------------------------------------
__global__ void t2_gemm(const float* __restrict__ q,
                        const float* __restrict__ kv,
                        float* __restrict__ v) {
    const int wave  = blockIdx.x * (blockDim.x >> 5) + (threadIdx.x >> 5);
    const int lane  = threadIdx.x & 31;
    const int laneN = lane & 15;
    const int half  = lane >> 4;

    const int JT = C_ / 64;       // 8
    const int MT = N_ / 32;       // 128
    const int b  = wave / (MT * JT);
    const int rem = wave % (MT * JT);
    const int m0 = (rem / JT) * 32;
    const int j0 = (rem % JT) * 64;

    const float* qb  = q  + (size_t)b * N_ * C_;
    const float* kvb = kv + (size_t)b * C_ * C_;

    v8f acc[2][4] = {};
#pragma unroll 2
    for (int k0 = 0; k0 < C_; k0 += 4) {
        const int kk = k0 + 2 * half;
        v2f a[2];
        a[0] = *(const v2f*)(qb + (size_t)(m0 + laneN) * C_ + kk);
        a[1] = *(const v2f*)(qb + (size_t)(m0 + 16 + laneN) * C_ + kk);
        const float* p0 = kvb + (size_t)kk * C_ + j0 + laneN;
        const float* p1 = p0 + C_;
#pragma unroll
        for (int t = 0; t < 4; ++t) {
            v2f bf;
            bf.x = p0[t * 16];
            bf.y = p1[t * 16];
            acc[0][t] = wmma4(a[0], bf, acc[0][t]);
            acc[1][t] = wmma4(a[1], bf, acc[1][t]);
        }
    }

    float* vb = v + (size_t)b * N_ * C_;
#pragma unroll
    for (int mt = 0; mt < 2; ++mt) {
#pragma unroll
        for (int t = 0; t < 4; ++t) {
#pragma unroll
            for (int r = 0; r < 8; ++r) {
                const int m = m0 + 16 * mt + r + 8 * half;
                float* p = vb + (size_t)m * C_ + j0 + t * 16 + laneN;
                *p = *p + acc[mt][t][r] * INV_SQRT_N;
            }
        }
    }
}

// ---------------------------------------------------------------------------
// K4b: out[b,c,n] = (attn @ Wln + bln) transposed-store via LDS (coalesced).
// One wave per block (LDS tile private, syncs are wave-local).
// ---------------------------------------------------------------------------
__global__ void out_gemm(const float* __restrict__ attn,
                         const float* __restrict__ Wln,
                         const float* __restrict__ bln,
                         float* __restrict__ out) {
    __shared__ float lds[16 * 17];
    const int wave  = blockIdx.x;
    const int lane  = threadIdx.x & 31;
    const int laneN = lane & 15;
    const int half  = lane >> 4;

    const int JT = C_ / 64;       // 8
    const int MT = N_ / 32;       // 128
    const int b  = wave / (MT * JT);
    const int rem = wave % (MT * JT);
    const int m0 = (rem / JT) * 32;
    const int j0 = (rem % JT) * 64;

    const float* ab = attn + (size_t)b * N_ * C_;

    v8f acc[2][4] = {};
#pragma unroll 2
    for (int k0 = 0; k0 < C_; k0 += 4) {
        const int kk = k0 + 2 * half;
        v2f a[2];
        a[0] = *(const v2f*)(ab + (size_t)(m0 + laneN) * C_ + kk);
        a[1] = *(const v2f*)(ab + (size_t)(m0 + 16 + laneN) * C_ + kk);
        const float* p0 = Wln + (size_t)kk * C_ + j0 + laneN;
        const float* p1 = p0 + C_;
#pragma unroll
        for (int t = 0; t < 4; ++t) {
            v2f bf;
            bf.x = p0[t * 16];
            bf.y = p1[t * 16];
            acc[0][t] = wmma4(a[0], bf, acc[0][t]);
            acc[1][t] = wmma4(a[1], bf, acc[1][t]);
        }
    }

    float* ob = out + (size_t)b * C_ * N_;
    for (int t = 0; t < 4; ++t) {
        const float bb = bln[j0 + t * 16 + laneN];
        for (int mt = 0; mt < 2; ++mt) {
#pragma unroll
            for (int r = 0; r < 8; ++r) {
                lds[(r + 8 * half) * 17 + laneN] = acc[mt][t][r] + bb;  // [n_local][c_local]
            }
            __syncthreads();
#pragma unroll
            for (int i = 0; i < 8; ++i) {
                const int idx = lane + 32 * i;   // 0..255
                const int cl  = idx >> 4;
                const int nl  = idx & 15;
                ob[(size_t)(j0 + t * 16 + cl) * N_ + m0 + 16 * mt + nl] = lds[nl * 17 + cl];
            }
            __syncthreads();
        }
    }
}

// ---------------------------------------------------------------------------
extern "C" void kernel_launch(void* const* d_in, const int* in_sizes, int n_in,
                              void* d_out, int out_size, void* d_ws, size_t ws_size,
                              hipStream_t stream) {
    const float* x    = (const float*)d_in[0];
    const float* Wqkv = (const float*)d_in[1];
    const float* bqkv = (const float*)d_in[2];
    const float* Wln  = (const float*)d_in[3];
    const float* bln  = (const float*)d_in[4];
    float* out = (float*)d_out;

    // Workspace layout (floats): q | k | v | kv | colnorm  (~210 MB)
    const size_t BNC = (size_t)B_ * N_ * C_;
    const size_t BCC = (size_t)B_ * C_ * C_;
    float* ws = (float*)d_ws;
    float* q  = ws;
    float* k  = ws + BNC;
    float* v  = ws + 2 * BNC;
    float* kv = ws + 3 * BNC;
    float* cn = ws + 3 * BNC + BCC;

    // K1: qkv GEMM. waves = B * (N/32) * (3C/64) = 24576; 4 waves/block.
    qkv_gemm<<<24576 / 4, 128, 0, stream>>>(x, Wqkv, bqkv, q, k, v);

    // K2: per-row normalization. one 256-thread block per (b,n) row.
    rownorm<<<B_ * N_, 256, 0, stream>>>(q, k);

    // K2b: column norms of k2_tmp.
    colnorm_kernel<<<B_ * (C_ / 256), 256, 0, stream>>>(k, cn);

    // K3: kv GEMM. waves = B * (C/32) * (C/64) = 1024; 4 waves/block.
    kv_gemm<<<1024 / 4, 128, 0, stream>>>(k, v, cn, kv);

    // K4a: t2 GEMM + residual into v. waves = B * (N/32) * (C/64) = 8192.
    t2_gemm<<<8192 / 4, 128, 0, stream>>>(q, kv, v);

    // K4b: final GEMM + transposed store. one wave per block.
    out_gemm<<<8192, 32, 0, stream>>>(v, Wln, bln, out);
}